// LengthRegulator_71416716198351
// MI455X (gfx1250) — compile-verified
//
#include <hip/hip_runtime.h>
#include <stdint.h>

// Length regulator: out[b,pos,:] = x[b, searchsorted(cum[b], pos, 'right'), :] * (pos < total[b])
// Geometry fixed by the reference:
#define LR_B      32
#define LR_N      256
#define LR_D      384
#define LR_MAXDUR 2048
#define LR_VECD   (LR_D / 4)   // 96 float4 per row = 3 per lane in a wave32

// ---------------------------------------------------------------------------
// Kernel 1: per-batch inclusive scan of durations, one wave32 per batch.
// Each lane: 8 contiguous ints (two b128 loads) -> local scan -> 5-step
// wave shuffle scan (DS permute path) -> two b128 stores. No barriers.
// Writes cum[b][n] to workspace and total[b] (as float) to d_out tail.
// ---------------------------------------------------------------------------
__global__ __launch_bounds__(32) void lr_scan_kernel(const int* __restrict__ dur,
                                                     int* __restrict__ cum,
                                                     float* __restrict__ total_out) {
  const int b    = blockIdx.x;
  const int lane = threadIdx.x;

  const int4* __restrict__ dv = (const int4*)(dur + (size_t)b * LR_N);
  int4 a0 = dv[lane * 2 + 0];
  int4 a1 = dv[lane * 2 + 1];

  int v[8] = {a0.x, a0.y, a0.z, a0.w, a1.x, a1.y, a1.z, a1.w};
#pragma unroll
  for (int i = 1; i < 8; ++i) v[i] += v[i - 1];

  // Wave32 inclusive scan of per-lane sums.
  int incl = v[7];
#pragma unroll
  for (int off = 1; off < 32; off <<= 1) {
    int n = __shfl_up(incl, off, 32);
    if (lane >= off) incl += n;
  }
  const int excl = incl - v[7];
#pragma unroll
  for (int i = 0; i < 8; ++i) v[i] += excl;

  int4 o0 = {v[0], v[1], v[2], v[3]};
  int4 o1 = {v[4], v[5], v[6], v[7]};
  int4* __restrict__ cv = (int4*)(cum + (size_t)b * LR_N);
  cv[lane * 2 + 0] = o0;
  cv[lane * 2 + 1] = o1;
  if (lane == 31) total_out[b] = (float)v[7];
}

// ---------------------------------------------------------------------------
// Kernel 2: gather/broadcast. grid = (B, MAXDUR/256), 256 threads = 8 wave32.
//  - cum table staged into LDS via gfx1250 async-to-LDS (ASYNCcnt path)
//  - one 8-step binary search per output row (LDS-resident, DS traffic only)
//  - each wave owns 32 CONTIGUOUS rows; source index is wave-uniform and
//    monotone, so the 3x float4 source row is kept in registers and only
//    reloaded on index change (durations 0..7 => ~3.5 stores per load)
//  - global_prefetch_b8 on the next source row at index transitions
// ---------------------------------------------------------------------------
__global__ __launch_bounds__(256) void lr_gather_kernel(const float* __restrict__ x,
                                                        const int* __restrict__ cum,
                                                        float* __restrict__ out) {
  __shared__ int s_cum[LR_N];
  __shared__ int s_idx[256];

  const int b    = blockIdx.x;
  const int row0 = blockIdx.y * 256;
  const int tid  = threadIdx.x;

  // Stage cum[b][0..255] into LDS with the CDNA5 async load-to-LDS path.
  // Generic pointers to __shared__ carry the workgroup LDS byte offset in
  // their low 32 bits (flat-aperture rule), matching the instruction's
  // LDS-address VGPR.
  {
    unsigned           lds_off = (unsigned)(uintptr_t)(&s_cum[tid]);
    unsigned long long gaddr   = (unsigned long long)(uintptr_t)(cum + b * LR_N + tid);
    asm volatile("global_load_async_to_lds_b32 %0, %1, off"
                 :
                 : "v"(lds_off), "v"(gaddr)
                 : "memory");
    asm volatile("s_wait_asynccnt 0" ::: "memory");
  }
  __syncthreads();

  const int total = s_cum[LR_N - 1];
  const int pos   = row0 + tid;

  // searchsorted(cum, pos, side='right'): first n with cum[n] > pos, in [0,N].
  int lo = 0, hi = LR_N;
#pragma unroll
  for (int it = 0; it < 8; ++it) {
    const int  mid = (lo + hi) >> 1;
    const bool le  = (s_cum[mid] <= pos);
    lo = le ? (mid + 1) : lo;
    hi = le ? hi : mid;
  }
  const int idxc = (lo < LR_N - 1) ? lo : (LR_N - 1);  // clip to N-1
  s_idx[tid] = (pos < total) ? idxc : -1;              // -1 => zero row
  __syncthreads();

  const int lane = tid & 31;
  const int wv   = tid >> 5;
  const int base = wv * 32;  // 32 contiguous rows per wave

  const float4* __restrict__ xv = (const float4*)x;
  float4* __restrict__       ov = (float4*)out;

  int    cur = -2;  // sentinel: forces load on first row
  float4 r0  = make_float4(0.f, 0.f, 0.f, 0.f), r1 = r0, r2 = r0;

  for (int r = 0; r < 32; ++r) {
    const int row = base + r;
    const int src = s_idx[row];  // wave-uniform
    if (src != cur) {
      if (src >= 0) {
        const float4* sp = xv + ((size_t)b * LR_N + src) * LR_VECD + lane;
        r0 = sp[0];
        r1 = sp[32];
        r2 = sp[64];
      } else {
        r0 = r1 = r2 = make_float4(0.f, 0.f, 0.f, 0.f);
      }
      cur = src;
      // Prefetch the next distinct source row (indices are monotone).
      if (row + 1 < base + 32) {
        const int ns = s_idx[row + 1];
        if (ns > src)
          __builtin_prefetch(x + ((size_t)b * LR_N + ns) * LR_D + lane * 12, 0, 0);
      }
    }
    float4* dst = ov + ((size_t)b * LR_MAXDUR + row0 + row) * LR_VECD + lane;
    dst[0]  = r0;
    dst[32] = r1;
    dst[64] = r2;
  }
}

// ---------------------------------------------------------------------------
// Launch
// ---------------------------------------------------------------------------
extern "C" void kernel_launch(void* const* d_in, const int* in_sizes, int n_in,
                              void* d_out, int out_size, void* d_ws, size_t ws_size,
                              hipStream_t stream) {
  (void)in_sizes; (void)n_in; (void)out_size; (void)ws_size;

  const float* x   = (const float*)d_in[0];  // (B, N, D) f32
  const int*   dur = (const int*)d_in[1];    // (B, N) i32
  // d_in[2] is the max_dur scalar (2048); geometry is compile-time here.

  float* out       = (float*)d_out;                          // (B, MAXDUR, D)
  float* total_out = out + (size_t)LR_B * LR_MAXDUR * LR_D;  // (B,) tail
  int*   cum       = (int*)d_ws;                             // B*N ints scratch

  lr_scan_kernel<<<LR_B, 32, 0, stream>>>(dur, cum, total_out);

  dim3 grid(LR_B, LR_MAXDUR / 256);
  lr_gather_kernel<<<grid, 256, 0, stream>>>(x, cum, out);
}